// CompressDCT_28587302322837
// MI455X (gfx1250) — compile-verified
//
#include <hip/hip_runtime.h>

typedef float v2f __attribute__((ext_vector_type(2)));
typedef float v8f __attribute__((ext_vector_type(8)));

#define WAVES_PER_BLOCK 8
#define TILES_PER_WAVE  4
#define LDS_PITCH       18  // 18*L mod 32 distinct for L=0..15; keeps b64 loads 8B-aligned

// Orthonormal DCT-II matrix D[k][n] = s_k * cos(pi*(2n+1)k/16), s_0=sqrt(1/8), s_k=0.5
__constant__ float Dmat[64] = {
    0.35355339059327373f,  0.35355339059327373f,  0.35355339059327373f,  0.35355339059327373f,
    0.35355339059327373f,  0.35355339059327373f,  0.35355339059327373f,  0.35355339059327373f,
    0.49039264020161522f,  0.41573480615127262f,  0.27778511650980114f,  0.09754516100806417f,
   -0.09754516100806417f, -0.27778511650980114f, -0.41573480615127262f, -0.49039264020161522f,
    0.46193976625564337f,  0.19134171618254492f, -0.19134171618254492f, -0.46193976625564337f,
   -0.46193976625564337f, -0.19134171618254492f,  0.19134171618254492f,  0.46193976625564337f,
    0.41573480615127262f, -0.09754516100806417f, -0.49039264020161522f, -0.27778511650980114f,
    0.27778511650980114f,  0.49039264020161522f,  0.09754516100806417f, -0.41573480615127262f,
    0.35355339059327379f, -0.35355339059327379f, -0.35355339059327379f,  0.35355339059327379f,
    0.35355339059327379f, -0.35355339059327379f, -0.35355339059327379f,  0.35355339059327379f,
    0.27778511650980114f, -0.49039264020161522f,  0.09754516100806417f,  0.41573480615127262f,
   -0.41573480615127262f, -0.09754516100806417f,  0.49039264020161522f, -0.27778511650980114f,
    0.19134171618254492f, -0.46193976625564337f,  0.46193976625564337f, -0.19134171618254492f,
   -0.19134171618254492f,  0.46193976625564337f, -0.46193976625564337f,  0.19134171618254492f,
    0.09754516100806417f, -0.27778511650980114f,  0.41573480615127262f, -0.49039264020161522f,
    0.49039264020161522f, -0.41573480615127262f,  0.27778511650980114f, -0.09754516100806417f
};

// M = blockdiag(D, D): zero when row/col are in different 8x8 blocks
__device__ __forceinline__ float mval(int i, int j) {
    float d = Dmat[((i & 7) << 3) | (j & 7)];
    return (((i ^ j) & 8) != 0) ? 0.0f : d;
}

__global__ void __launch_bounds__(256)
dct_quant_kernel(const float* __restrict__ x,
                 const float* __restrict__ qtab,
                 const int*   __restrict__ bitp,
                 float*       __restrict__ out)
{
    constexpr int W  = 128;        // reference shape (32, 64, 128, 128)
    constexpr int HW = 128 * 128;

    __shared__ float lds[WAVES_PER_BLOCK][2][16 * LDS_PITCH];

    const int waveInBlk = threadIdx.x >> 5;
    const int lane      = threadIdx.x & 31;
    const int halfSel   = lane >> 4;     // 0: lanes 0-15, 1: lanes 16-31
    const int col16     = lane & 15;

    // 4 consecutive tiles along one 16-row band (64 tiles/slice, 8 tiles/row)
    const int wave  = blockIdx.x * WAVES_PER_BLOCK + waveInBlk;
    const int t0    = wave * TILES_PER_WAVE;
    const int slice = t0 >> 6;
    const int rem   = t0 & 63;
    const int ty    = rem >> 3;
    const int tx0   = rem & 7;
    const size_t rowBase = (size_t)slice * HW + (size_t)(ty * 16) * W + (size_t)(tx0 * 16);

    // ---- constant DCT chunks: A-layout of M == B-layout of M^T (hoisted, table-driven) ----
    v2f mChunk[4];
    #pragma unroll
    for (int c = 0; c < 4; ++c) {
        const int kk = 4 * c + 2 * halfSel;
        mChunk[c].x = mval(col16, kk);
        mChunk[c].y = mval(col16, kk + 1);
    }

    // ---- per-lane q_table reciprocals (exact for q=1); amortized over 4 tiles ----
    float rq[8];
    #pragma unroll
    for (int v = 0; v < 8; ++v)
        rq[v] = 1.0f / qtab[v * 8 + (col16 & 7)];

    const int   bit = bitp[0];
    const float hi  = (float)((1 << (bit - 1)) - 1);
    const float lo  = -(float)(1 << (bit - 1));

    #pragma unroll
    for (int t = 0; t < TILES_PER_WAVE; ++t) {
        const float* __restrict__ xb = x   + rowBase + t * 16;
        float*       __restrict__ ob = out + rowBase + t * 16;

        // ---- matmul 1: T = M @ X ; X loaded straight into B-layout ----
        v8f acc = {0.f, 0.f, 0.f, 0.f, 0.f, 0.f, 0.f, 0.f};
        #pragma unroll
        for (int c = 0; c < 4; ++c) {
            const int rb = 4 * c + 2 * halfSel;
            v2f b;
            b.x = xb[rb * W + col16];
            b.y = xb[(rb + 1) * W + col16];
            acc = __builtin_amdgcn_wmma_f32_16x16x4_f32(
                      false, mChunk[c], false, b, (short)0, acc, false, false);
        }

        // ---- stage T through double-buffered per-wave LDS (D-layout -> A-layout) ----
        float* tl = lds[waveInBlk][t & 1];
        #pragma unroll
        for (int v = 0; v < 8; ++v)
            tl[(v + 8 * halfSel) * LDS_PITCH + col16] = acc[v];

        __syncthreads();   // orders the cross-lane transpose; buffers alternate -> one barrier

        // ---- matmul 2: Y = T @ M^T ; A = T chunks from LDS, B = same constants ----
        v8f acc2 = {0.f, 0.f, 0.f, 0.f, 0.f, 0.f, 0.f, 0.f};
        #pragma unroll
        for (int c = 0; c < 4; ++c) {
            const int kk = 4 * c + 2 * halfSel;
            const float2 a2 = *(const float2*)&tl[col16 * LDS_PITCH + kk]; // ds_load_b64
            v2f a; a.x = a2.x; a.y = a2.y;
            acc2 = __builtin_amdgcn_wmma_f32_16x16x4_f32(
                       false, a, false, mChunk[c], (short)0, acc2, false, false);
        }

        // ---- quantize: round-half-even, clamp, store in D-layout positions ----
        #pragma unroll
        for (int v = 0; v < 8; ++v) {
            float q = rintf(acc2[v] * rq[v]);   // v_rndne_f32 == jnp.round
            q = fminf(fmaxf(q, lo), hi);
            ob[(v + 8 * halfSel) * W + col16] = q;
        }
    }
}

extern "C" void kernel_launch(void* const* d_in, const int* in_sizes, int n_in,
                              void* d_out, int out_size, void* d_ws, size_t ws_size,
                              hipStream_t stream) {
    const float* x    = (const float*)d_in[0];
    const float* qtab = (const float*)d_in[1];
    const int*   bit  = (const int*)d_in[2];
    float*       out  = (float*)d_out;

    const int nTiles = in_sizes[0] / 256;                       // 131072
    const int waves  = nTiles / TILES_PER_WAVE;                 // 32768
    const int blocks = waves / WAVES_PER_BLOCK;                 // 4096

    dct_quant_kernel<<<blocks, WAVES_PER_BLOCK * 32, 0, stream>>>(x, qtab, bit, out);
}